// TensorRepsTransform_63995012710862
// MI455X (gfx1250) — compile-verified
//
#include <hip/hip_runtime.h>

// CDNA5 / gfx1250: wave32, WMMA f32 16x16x4 (A 16x4, B 4x16, C/D 16x16).
typedef float v2f __attribute__((ext_vector_type(2)));
typedef float v8f __attribute__((ext_vector_type(8)));

#define DIM 832
#define WAVES_PER_BLOCK 8

// Exact branchless 2-way select: returns (h ? hi : lo). Lowers to v_bfi/cndmask.
__device__ __forceinline__ float lane_sel(int h, float lo, float hi) {
  unsigned m = 0u - (unsigned)h;                  // 0 or 0xFFFFFFFF
  unsigned r = (__float_as_uint(lo) & ~m) | (__float_as_uint(hi) & m);
  return __uint_as_float(r);
}

// One sample per wave32.  D = A x B with
//   A (16x4) = Lambda replicated 4x vertically: A[m,k] = L[(m&3)*4 + k]
//   B (4x16) = 16 data 4-vectors in columns:    B[k,n] = v_n[k]
// so for every lane, acc[g] = (Lambda v_col)[g], g = 0..3 (rows 4..15 replicate).
//
// Operand layouts (CDNA5 ISA 7.12.2, 32-bit):
//   A : lane l -> VGPR v = A[l&15, 2*(l>=16)+v]
//   B : lane l -> VGPR v = B[2*(l>=16)+v, l&15]
//   C/D: VGPR g, lanes 0-15 -> D[g, lane]; lanes 16-31 -> D[g+8, lane-16]
__global__ __launch_bounds__(256) void TensorRepsTransform_kernel(
    const float* __restrict__ tensor,   // (N, 832)
    const float* __restrict__ lframes,  // (N, 4, 4)
    float* __restrict__ out,            // (N, 832)
    int n)
{
  __shared__ __align__(16) float lds_all[WAVES_PER_BLOCK][512]; // 128 vecs x 4 per wave

  const int tid  = (int)threadIdx.x;
  const int lane = tid & 31;
  const int wid  = tid >> 5;
  const int half = lane >> 4;   // K-pair select for A/B loads; component-pair for D
  const int col  = lane & 15;   // which vector (B column) this lane owns
  const int j    = col & 3;

  const int s = (int)blockIdx.x * WAVES_PER_BLOCK + wid;
  if (s >= n) return;           // whole-wave uniform; EXEC all-ones for WMMA

  const float* __restrict__ x = tensor  + (long long)s * DIM;
  const float* __restrict__ L = lframes + (long long)s * 16;
  float*       __restrict__ y = out     + (long long)s * DIM;
  float* lds = lds_all[wid];

  // ---- A = Lambda replicated: lane holds A[lane&15, 2*half + v] = L[j*4 + 2*half + v]
  const v2f Am = *(const v2f*)(L + j * 4 + 2 * half);

  // ---- sign(det Lambda), cofactor expansion (all lanes redundantly) ----
  const float l00=L[0],  l01=L[1],  l02=L[2],  l03=L[3];
  const float l10=L[4],  l11=L[5],  l12=L[6],  l13=L[7];
  const float l20=L[8],  l21=L[9],  l22=L[10], l23=L[11];
  const float l30=L[12], l31=L[13], l32=L[14], l33=L[15];
  const float cA0=l00*l11-l01*l10, cA1=l00*l12-l02*l10, cA2=l00*l13-l03*l10;
  const float cA3=l01*l12-l02*l11, cA4=l01*l13-l03*l11, cA5=l02*l13-l03*l12;
  const float cB0=l20*l31-l21*l30, cB1=l20*l32-l22*l30, cB2=l20*l33-l23*l30;
  const float cB3=l21*l32-l22*l31, cB4=l21*l33-l23*l31, cB5=l22*l33-l23*l32;
  const float det = cA0*cB5 - cA1*cB4 + cA2*cB3 + cA3*cB2 - cA4*cB1 + cA5*cB0;
  const float sgn = (det > 0.f) ? 1.f : ((det < 0.f) ? -1.f : 0.f);

  // ---- order-0 passthrough (64 floats, coalesced b64) ----
  *(v2f*)(y + 2 * lane) = *(const v2f*)(x + 2 * lane);

  // ---- pass 1: channels [64,832) are 192 contiguous float4 vectors; v' = Lambda v
  // batches 0-3  -> order-1 vectors (scale by sgn, coalesced b64 store to global)
  // batches 4-11 -> order-2 tensor rows X_i = T_i*Lambda^T... stored transposed to LDS
  for (int b = 0; b < 12; ++b) {
    const int r = 16 * b + col;                       // vector index 0..191
    // B[k=2*half+v, col] = v_r[2*half+v]; 256B contiguous per instruction
    const v2f Bm = *(const v2f*)(x + 64 + 4 * r + 2 * half);
    v8f acc = {0.f, 0.f, 0.f, 0.f, 0.f, 0.f, 0.f, 0.f};
    acc = __builtin_amdgcn_wmma_f32_16x16x4_f32(
        /*neg_a=*/false, Am, /*neg_b=*/false, Bm,
        /*c_mod=*/(short)0, acc, /*reuse_a=*/false, /*reuse_b=*/false);
    // lane's components: j0 = 2*half, j1 = 2*half+1 (single branchless select)
    const float s0 = lane_sel(half, acc[0], acc[2]);
    const float s1 = lane_sel(half, acc[1], acc[3]);
    if (b < 4) {
      // order-1 (parity-odd): out[64 + 4r + {2h, 2h+1}], 32 lanes -> 256B contiguous
      v2f o; o.x = sgn * s0; o.y = sgn * s1;
      *(v2f*)(y + 64 + 4 * r + 2 * half) = o;
    } else {
      // value = X_i[k, comp]; stage as lds2[4*i + comp][k] (rows of X^T for pass 2)
      const int idx  = r - 64;                        // 0..127
      const int base = (idx & ~3) * 4 + (idx & 3) + 8 * half;
      lds[base]     = s0;                             // comp 2*half
      lds[base + 4] = s1;                             // comp 2*half+1
    }
  }

  // LDS ops are in-order per wave; fence the compiler + drain DScnt before cross-lane reads.
  asm volatile("s_wait_dscnt 0" ::: "memory");

  // ---- pass 2: w_q = row q of X^T; (Lambda w_q)[jj] = T'[jj, q&3] ----
  for (int b = 0; b < 8; ++b) {
    const int q = 16 * b + col;                       // 0..127
    const v2f Bm = *(const v2f*)(lds + 4 * q + 2 * half);  // conflict-free b64
    v8f acc = {0.f, 0.f, 0.f, 0.f, 0.f, 0.f, 0.f, 0.f};
    acc = __builtin_amdgcn_wmma_f32_16x16x4_f32(
        false, Am, false, Bm, (short)0, acc, false, false);
    const float s0 = lane_sel(half, acc[0], acc[2]);
    const float s1 = lane_sel(half, acc[1], acc[3]);
    // T'[jj, m] of matrix i=q>>2, m=q&3 -> out[320 + 16*i + 4*jj + m], jj = 2h, 2h+1
    const int base = 320 + (q & ~3) * 4 + (q & 3) + 8 * half;
    y[base]     = s0;
    y[base + 4] = s1;
  }
}

extern "C" void kernel_launch(void* const* d_in, const int* in_sizes, int n_in,
                              void* d_out, int out_size, void* d_ws, size_t ws_size,
                              hipStream_t stream) {
  const float* tensor  = (const float*)d_in[0];   // (N, 832) f32
  const float* lframes = (const float*)d_in[1];   // (N, 4, 4) f32
  // d_in[2] = parity_odd mask; the rep structure (order-1 block odd) is hardcoded.
  float* out = (float*)d_out;

  const int n = in_sizes[0] / DIM;                // N samples
  const int blocks = (n + WAVES_PER_BLOCK - 1) / WAVES_PER_BLOCK;
  TensorRepsTransform_kernel<<<blocks, WAVES_PER_BLOCK * 32, 0, stream>>>(
      tensor, lframes, out, n);
}